// GeoEncoder_86071144612563
// MI455X (gfx1250) — compile-verified
//
#include <hip/hip_runtime.h>

// GeoEncoder (TensoRF-style tri-plane + line) forward for MI455X / gfx1250.
//
// Roofline: planes (151 MB) are L2-resident (192 MB L2); output (128 MB) and
// coordinates (12 MB) are streamed with non-temporal hints so they do not
// evict the planes. The kernel is L2-gather bound (~576M 4B gathers); the
// 1e6 x 48 x 32 projection GEMM runs on V_WMMA_F32_16X16X4_F32 (fp32, matching
// the fp32 reference), 48 WMMAs per 32-point wave tile.

#define RES   512
#define RES2  (RES * RES)
#define RANK  48
#define OUTC  32
#define KSTEPS (RANK / 4)          // 12 WMMA K-steps of 4
#define WAVES_PER_BLOCK 8
#define BLOCK_THREADS   256
#define LDS_STRIDE 50              // RANK padded (even -> b64-aligned, bank-friendly)

typedef __attribute__((ext_vector_type(2))) float v2f;
typedef __attribute__((ext_vector_type(8))) float v8f;

__device__ __forceinline__ void axis_setup(float c, int& i0, int& i1, float& w) {
    float p = fminf(fmaxf(c * (float)(RES - 1), 0.0f), (float)(RES - 1));
    float f = floorf(p);
    i0 = (int)f;
    i1 = min(i0 + 1, RES - 1);
    w = p - f;
}

__device__ __forceinline__ float lerp1(float a, float b, float w) {
    return fmaf(w, b - a, a);
}

__global__ __launch_bounds__(BLOCK_THREADS)
void geo_encoder_kernel(const float* __restrict__ coords,
                        const float* __restrict__ plane_xy,
                        const float* __restrict__ plane_xz,
                        const float* __restrict__ plane_yz,
                        const float* __restrict__ line_z,
                        const float* __restrict__ line_y,
                        const float* __restrict__ line_x,
                        const float* __restrict__ proj_w,
                        const float* __restrict__ proj_b,
                        float* __restrict__ out,
                        int n_pts)
{
    __shared__ float vmLds[WAVES_PER_BLOCK][32][LDS_STRIDE];

    const int lane     = threadIdx.x & 31;
    const int wave     = threadIdx.x >> 5;
    const int tileBase = (blockIdx.x * WAVES_PER_BLOCK + wave) * 32;
    const int halfSel  = lane >> 4;   // 0: lanes 0-15, 1: lanes 16-31
    const int nLane    = lane & 15;

    const int pt  = tileBase + lane;
    const int ptc = min(pt, n_pts - 1);   // clamp for tail tile; stores are masked

    // ---- per-point setup: indices + bilinear weights (streaming coord load) ----
    float cx = __builtin_nontemporal_load(coords + ptc * 3 + 0);
    float cy = __builtin_nontemporal_load(coords + ptc * 3 + 1);
    float cz = __builtin_nontemporal_load(coords + ptc * 3 + 2);

    int ix0, ix1, iy0, iy1, iz0, iz1;
    float wx, wy, wz;
    axis_setup(cx, ix0, ix1, wx);
    axis_setup(cy, iy0, iy1, wy);
    axis_setup(cz, iz0, iz1, wz);
    const float ux = 1.0f - wx, uy = 1.0f - wy, uz = 1.0f - wz;

    // plane_xy sampled at (x<-cx, y<-cy); plane_xz at (x<-cx, y<-cz); plane_yz at (x<-cy, y<-cz)
    const float wxy00 = uy * ux, wxy01 = uy * wx, wxy10 = wy * ux, wxy11 = wy * wx;
    const float wxz00 = uz * ux, wxz01 = uz * wx, wxz10 = wz * ux, wxz11 = wz * wx;
    const float wyz00 = uz * uy, wyz01 = uz * wy, wyz10 = wz * uy, wyz11 = wz * wy;

    const int oxy00 = iy0 * RES + ix0, oxy01 = iy0 * RES + ix1;
    const int oxy10 = iy1 * RES + ix0, oxy11 = iy1 * RES + ix1;
    const int oxz00 = iz0 * RES + ix0, oxz01 = iz0 * RES + ix1;
    const int oxz10 = iz1 * RES + ix0, oxz11 = iz1 * RES + ix1;
    const int oyz00 = iz0 * RES + iy0, oyz01 = iz0 * RES + iy1;
    const int oyz10 = iz1 * RES + iy0, oyz11 = iz1 * RES + iy1;

    // ---- preload B fragments (proj_w, 6 KB, L2-resident) ----
    // B[k][n] = proj_w[n][k]; lane l supplies n = nLane (+16*nt),
    // k = 4*ks + 2*halfSel + {0,1}  -> one float2 per (nt, ks).
    v2f bfrag[2][KSTEPS];
#pragma unroll
    for (int nt = 0; nt < 2; ++nt) {
#pragma unroll
        for (int ks = 0; ks < KSTEPS; ++ks) {
            const float* p = proj_w + (nt * 16 + nLane) * RANK + ks * 4 + 2 * halfSel;
            v2f b; b.x = p[0]; b.y = p[1];
            bfrag[nt][ks] = b;
        }
    }

    // ---- accumulators: init with bias (broadcast along M; bias depends on N only) ----
    const float bias0 = proj_b[nLane];
    const float bias1 = proj_b[16 + nLane];
    v8f acc[2][2];
#pragma unroll
    for (int mt = 0; mt < 2; ++mt)
#pragma unroll
        for (int e = 0; e < 8; ++e) { acc[mt][0][e] = bias0; acc[mt][1][e] = bias1; }

    // ---- rank loop: 12 plane gathers + 6 line loads per r, all L2-resident ----
#pragma unroll 2
    for (int r = 0; r < RANK; ++r) {
        const float* pxy = plane_xy + (size_t)r * RES2;
        const float* pxz = plane_xz + (size_t)r * RES2;
        const float* pyz = plane_yz + (size_t)r * RES2;
        const float* lzp = line_z + r * RES;
        const float* lyp = line_y + r * RES;
        const float* lxp = line_x + r * RES;

        float sxy = pxy[oxy00] * wxy00 + pxy[oxy01] * wxy01
                  + pxy[oxy10] * wxy10 + pxy[oxy11] * wxy11;
        float sxz = pxz[oxz00] * wxz00 + pxz[oxz01] * wxz01
                  + pxz[oxz10] * wxz10 + pxz[oxz11] * wxz11;
        float syz = pyz[oyz00] * wyz00 + pyz[oyz01] * wyz01
                  + pyz[oyz10] * wyz10 + pyz[oyz11] * wyz11;

        float lz = lerp1(lzp[iz0], lzp[iz1], wz);
        float ly = lerp1(lyp[iy0], lyp[iy1], wy);
        float lx = lerp1(lxp[ix0], lxp[ix1], wx);

        vmLds[wave][lane][r] = sxy * lz + sxz * ly + syz * lx;
    }

    // Same-wave DS ops are hardware in-order; explicit CDNA5 wait as a fence
    // before cross-lane fragment reads.
    asm volatile("s_wait_dscnt 0x0" ::: "memory");

    // ---- WMMA projection: D[32x32] = vm[32x48] * W^T[48x32]  (fp32) ----
    // A 16x4 f32 fragment: lane l -> row l%16 (+16*mt), VGPR0/1 = K {0,1} or {2,3}
    // selected by halfSel -> contiguous float2 from LDS.
    const float* aBase = &vmLds[wave][0][0];
#pragma unroll
    for (int ks = 0; ks < KSTEPS; ++ks) {
#pragma unroll
        for (int mt = 0; mt < 2; ++mt) {
            const float* ap = aBase + (mt * 16 + nLane) * LDS_STRIDE + ks * 4 + 2 * halfSel;
            v2f afrag; afrag.x = ap[0]; afrag.y = ap[1];
            acc[mt][0] = __builtin_amdgcn_wmma_f32_16x16x4_f32(
                false, afrag, false, bfrag[0][ks], (short)0, acc[mt][0], false, false);
            acc[mt][1] = __builtin_amdgcn_wmma_f32_16x16x4_f32(
                false, afrag, false, bfrag[1][ks], (short)0, acc[mt][1], false, false);
        }
    }

    // ---- masked, non-temporal store (keep 128 MB output stream out of L2) ----
    // C/D layout: VGPR v holds row v (lanes 0-15) / v+8 (lanes 16-31), col = l%16.
#pragma unroll
    for (int mt = 0; mt < 2; ++mt) {
#pragma unroll
        for (int v = 0; v < 8; ++v) {
            int row = tileBase + mt * 16 + halfSel * 8 + v;
            if (row < n_pts) {
                __builtin_nontemporal_store(acc[mt][0][v], out + (size_t)row * OUTC + nLane);
                __builtin_nontemporal_store(acc[mt][1][v], out + (size_t)row * OUTC + 16 + nLane);
            }
        }
    }
}

extern "C" void kernel_launch(void* const* d_in, const int* in_sizes, int n_in,
                              void* d_out, int out_size, void* d_ws, size_t ws_size,
                              hipStream_t stream) {
    const float* coords   = (const float*)d_in[0];
    const float* plane_xy = (const float*)d_in[1];
    const float* plane_xz = (const float*)d_in[2];
    const float* plane_yz = (const float*)d_in[3];
    const float* line_z   = (const float*)d_in[4];
    const float* line_y   = (const float*)d_in[5];
    const float* line_x   = (const float*)d_in[6];
    const float* proj_w   = (const float*)d_in[7];
    const float* proj_b   = (const float*)d_in[8];
    float* out = (float*)d_out;

    const int n_pts = in_sizes[0] / 3;
    const int pts_per_block = BLOCK_THREADS; // 8 waves x 32 points
    const int blocks = (n_pts + pts_per_block - 1) / pts_per_block;

    geo_encoder_kernel<<<blocks, BLOCK_THREADS, 0, stream>>>(
        coords, plane_xy, plane_xz, plane_yz, line_z, line_y, line_x,
        proj_w, proj_b, out, n_pts);
}